// ChebNet_9294309229065
// MI455X (gfx1250) — compile-verified
//
#include <hip/hip_runtime.h>
#include <math.h>

// ---------------------------------------------------------------------------
// ChebNet (K=3) on gfx1250 / MI455X.
//  - Graph props: gather/scatter with global_atomic_add_f32 (L2-resident,
//    25.6MB working set << 192MB L2).
//  - GEMMs: V_WMMA_F32_16X16X4_F32 (fp32 exact, matches reference numerics),
//    weights staged in LDS (48KB max << 320KB/WGP), concat handled virtually
//    via 3 segment pointers.
// Wave32 everywhere; blocks of 256 threads = 8 waves.
// ---------------------------------------------------------------------------

typedef __attribute__((ext_vector_type(2))) float v2f;
typedef __attribute__((ext_vector_type(8))) float v8f;

#define FDIM 64  // IN_F == HID == 64 (per concat segment)

__global__ void zero_kernel(float* __restrict__ p, long long n) {
  long long i = (long long)blockIdx.x * blockDim.x + threadIdx.x;
  if (i < n) p[i] = 0.0f;
}

__global__ void deg_kernel(const int* __restrict__ dst, float* __restrict__ deg, int E) {
  int e = blockIdx.x * blockDim.x + threadIdx.x;
  if (e < E) atomicAdd(&deg[dst[e]], 1.0f);
}

__global__ void dinv_kernel(float* __restrict__ dinv, int N) {
  int n = blockIdx.x * blockDim.x + threadIdx.x;
  if (n < N) {
    float d = dinv[n];
    d = d < 1.0f ? 1.0f : d;          // clip(deg, 1, None)
    dinv[n] = 1.0f / sqrtf(d);        // deg^{-1/2}
  }
}

// hs[n][f] = x[n][f] * dinv[n]   (float4 per thread, 16 threads per node)
__global__ void scale_kernel(const float* __restrict__ x, const float* __restrict__ dinv,
                             float* __restrict__ out, int N) {
  int t = blockIdx.x * blockDim.x + threadIdx.x;
  int n = t >> 4;
  if (n >= N) return;
  int f = (t & 15) * 4;
  float s = dinv[n];
  const float4 v = *(const float4*)(x + (size_t)n * FDIM + f);
  float4 r; r.x = v.x * s; r.y = v.y * s; r.z = v.z * s; r.w = v.w * s;
  *(float4*)(out + (size_t)n * FDIM + f) = r;
}

// acc[dst[e]][:] += hs[src[e]][:]   (16 threads per edge, float4 each)
__global__ void scatter_kernel(const float* __restrict__ hs, const int* __restrict__ src,
                               const int* __restrict__ dst, float* __restrict__ acc, int E) {
  int t = blockIdx.x * blockDim.x + threadIdx.x;
  int e = t >> 4;
  if (e >= E) return;
  int f = (t & 15) * 4;
  int s = src[e], d = dst[e];
  const float4 v = *(const float4*)(hs + (size_t)s * FDIM + f);
  float* p = acc + (size_t)d * FDIM + f;
  atomicAdd(p + 0, v.x);
  atomicAdd(p + 1, v.y);
  atomicAdd(p + 2, v.z);
  atomicAdd(p + 3, v.w);
}

// out = alpha * acc * dinv[n]  + beta * x0     (X1: alpha=-1,beta=0; X2: alpha=-2,beta=-1)
__global__ void cheb_finish_kernel(const float* __restrict__ acc, const float* __restrict__ dinv,
                                   const float* __restrict__ x0, float* __restrict__ out,
                                   float alpha, float beta, int N) {
  int t = blockIdx.x * blockDim.x + threadIdx.x;
  int n = t >> 4;
  if (n >= N) return;
  int f = (t & 15) * 4;
  float s = alpha * dinv[n];
  const float4 a = *(const float4*)(acc + (size_t)n * FDIM + f);
  float4 r; r.x = a.x * s; r.y = a.y * s; r.z = a.z * s; r.w = a.w * s;
  if (beta != 0.0f) {  // grid-uniform branch
    const float4 x = *(const float4*)(x0 + (size_t)n * FDIM + f);
    r.x += beta * x.x; r.y += beta * x.y; r.z += beta * x.z; r.w += beta * x.w;
  }
  *(float4*)(out + (size_t)n * FDIM + f) = r;
}

// ---------------------------------------------------------------------------
// out[nRows x Ncols] = act( [A0|A1|..A(numSeg-1)] (nRows x numSeg*64) @ W + b )
// using V_WMMA_F32_16X16X4_F32. W (numSeg*64 x Ncols, row-major) staged in LDS.
// Block = 256 threads = 8 waves; each wave owns one 16x16 output tile.
// A-frag (16x4 f32): lane l -> row m=l&15, K pair at kb=(l>>4)*2.
// B-frag (4x16 f32): lane l -> col n=l&15, K pair at kb=(l>>4)*2.
// C/D: VGPR v -> row m=v+8*(l>>4), col n=l&15.
// Requires nRows % 16 == 0 (100000/16 = 6250, exact).
// ---------------------------------------------------------------------------
__global__ void wmma_gemm_kernel(const float* __restrict__ A0, const float* __restrict__ A1,
                                 const float* __restrict__ A2, int numSeg,
                                 const float* __restrict__ W, const float* __restrict__ bias,
                                 float* __restrict__ out, int nRows, int Ncols, int reluFlag) {
  extern __shared__ float sW[];
  const int Ktot = numSeg * FDIM;
  for (int i = threadIdx.x; i < Ktot * Ncols; i += blockDim.x) sW[i] = W[i];
  __syncthreads();

  const int lane = threadIdx.x & 31;
  const int wid  = threadIdx.x >> 5;
  const int tilesPerRow     = Ncols >> 4;        // 4 (Ncols=64) or 2 (Ncols=32)
  const int rowTilesPerBlk  = 8 / tilesPerRow;   // 2 or 4
  const int colTile = wid % tilesPerRow;
  const int rowTile = blockIdx.x * rowTilesPerBlk + wid / tilesPerRow;

  if (rowTile * 16 < nRows) {                    // wave-uniform: EXEC all-1s inside
    const int mLo = lane & 15;
    const int kb  = (lane >> 4) * 2;
    const int rowBase = rowTile * 16;
    const int colBase = colTile * 16;

    v8f acc = {};
    for (int seg = 0; seg < numSeg; ++seg) {
      const float* A = (seg == 0) ? A0 : (seg == 1) ? A1 : A2;
      const float* arow = A + (size_t)(rowBase + mLo) * FDIM;
      const float* wcol = sW + (size_t)seg * FDIM * Ncols + colBase + mLo;
#pragma unroll
      for (int k4 = 0; k4 < FDIM; k4 += 4) {
        const int k = k4 + kb;
        v2f a, b;
        a.x = arow[k];
        a.y = arow[k + 1];
        b.x = wcol[(size_t)k * Ncols];
        b.y = wcol[(size_t)(k + 1) * Ncols];
        acc = __builtin_amdgcn_wmma_f32_16x16x4_f32(
            /*neg_a=*/false, a, /*neg_b=*/false, b,
            /*c_mod=*/(short)0, acc, /*reuse_a=*/false, /*reuse_b=*/false);
      }
    }

    const float bv = bias[colBase + mLo];
    const int hi8 = (lane >> 4) * 8;
#pragma unroll
    for (int v = 0; v < 8; ++v) {
      float val = acc[v] + bv;
      if (reluFlag) val = fmaxf(val, 0.0f);
      out[(size_t)(rowBase + v + hi8) * Ncols + colBase + mLo] = val;
    }
  }
}

// ---------------------------------------------------------------------------

static inline int nblk(long long n, int t) { return (int)((n + t - 1) / t); }

extern "C" void kernel_launch(void* const* d_in, const int* in_sizes, int n_in,
                              void* d_out, int out_size, void* d_ws, size_t ws_size,
                              hipStream_t stream) {
  const float* features = (const float*)d_in[0];
  const int*   src      = (const int*)d_in[1];
  const int*   dst      = (const int*)d_in[2];
  // d_in[3] = n_nodes scalar (derived from sizes instead)
  const float* W1  = (const float*)d_in[4];
  const float* b1  = (const float*)d_in[5];
  const float* W2  = (const float*)d_in[6];
  const float* b2  = (const float*)d_in[7];
  const float* Wm1 = (const float*)d_in[8];
  const float* bm1 = (const float*)d_in[9];
  const float* Wm2 = (const float*)d_in[10];
  const float* bm2 = (const float*)d_in[11];

  const int N = in_sizes[0] / FDIM;   // 100000
  const int E = in_sizes[1];          // 1000000
  const long long NF = (long long)N * FDIM;

  // Workspace layout (floats): dinv | hs | acc | x1 | x2 | y1   (~128 MB)
  float* dinv = (float*)d_ws;
  float* hs   = dinv + N;
  float* acc  = hs + NF;
  float* x1   = acc + NF;
  float* x2   = x1 + NF;
  float* y1   = x2 + NF;
  float* y2   = acc;   // reuse: acc is free once layer-2 props are done
  float* h    = hs;    // reuse: hs is free during the MLP

  const long long NT = (long long)N * 16;  // threads for per-node float4 kernels
  const long long ET = (long long)E * 16;  // threads for per-edge float4 kernels

  // Degree -> dinv
  zero_kernel<<<nblk(N, 256), 256, 0, stream>>>(dinv, N);
  deg_kernel<<<nblk(E, 256), 256, 0, stream>>>(dst, dinv, E);
  dinv_kernel<<<nblk(N, 256), 256, 0, stream>>>(dinv, N);

  const int rowTiles = N / 16;  // 6250 (exact)

  // ---------------- Layer 1 (X0 = features) ----------------
  scale_kernel<<<nblk(NT, 256), 256, 0, stream>>>(features, dinv, hs, N);
  zero_kernel<<<nblk(NF, 256), 256, 0, stream>>>(acc, NF);
  scatter_kernel<<<nblk(ET, 256), 256, 0, stream>>>(hs, src, dst, acc, E);
  cheb_finish_kernel<<<nblk(NT, 256), 256, 0, stream>>>(acc, dinv, features, x1, -1.0f, 0.0f, N);

  scale_kernel<<<nblk(NT, 256), 256, 0, stream>>>(x1, dinv, hs, N);
  zero_kernel<<<nblk(NF, 256), 256, 0, stream>>>(acc, NF);
  scatter_kernel<<<nblk(ET, 256), 256, 0, stream>>>(hs, src, dst, acc, E);
  cheb_finish_kernel<<<nblk(NT, 256), 256, 0, stream>>>(acc, dinv, features, x2, -2.0f, -1.0f, N);

  {  // y1 = relu([X0|X1|X2] @ W1 + b1)
    const int Ncols = 64, rtpb = 8 / (Ncols >> 4);
    wmma_gemm_kernel<<<nblk(rowTiles, rtpb), 256, (size_t)3 * FDIM * Ncols * sizeof(float), stream>>>(
        features, x1, x2, 3, W1, b1, y1, N, Ncols, 1);
  }

  // ---------------- Layer 2 (X0 = y1) ----------------
  scale_kernel<<<nblk(NT, 256), 256, 0, stream>>>(y1, dinv, hs, N);
  zero_kernel<<<nblk(NF, 256), 256, 0, stream>>>(acc, NF);
  scatter_kernel<<<nblk(ET, 256), 256, 0, stream>>>(hs, src, dst, acc, E);
  cheb_finish_kernel<<<nblk(NT, 256), 256, 0, stream>>>(acc, dinv, y1, x1, -1.0f, 0.0f, N);

  scale_kernel<<<nblk(NT, 256), 256, 0, stream>>>(x1, dinv, hs, N);
  zero_kernel<<<nblk(NF, 256), 256, 0, stream>>>(acc, NF);
  scatter_kernel<<<nblk(ET, 256), 256, 0, stream>>>(hs, src, dst, acc, E);
  cheb_finish_kernel<<<nblk(NT, 256), 256, 0, stream>>>(acc, dinv, y1, x2, -2.0f, -1.0f, N);

  {  // y2 = relu([y1|x1|x2] @ W2 + b2)   (written into acc, which is now free)
    const int Ncols = 64, rtpb = 8 / (Ncols >> 4);
    wmma_gemm_kernel<<<nblk(rowTiles, rtpb), 256, (size_t)3 * FDIM * Ncols * sizeof(float), stream>>>(
        y1, x1, x2, 3, W2, b2, y2, N, Ncols, 1);
  }

  // ---------------- MLP ----------------
  {  // h = relu(y2 @ Wm1 + bm1)
    const int Ncols = 64, rtpb = 8 / (Ncols >> 4);
    wmma_gemm_kernel<<<nblk(rowTiles, rtpb), 256, (size_t)FDIM * Ncols * sizeof(float), stream>>>(
        y2, y2, y2, 1, Wm1, bm1, h, N, Ncols, 1);
  }
  {  // out = h @ Wm2 + bm2   (Ncols=32, partial last block handled by tile guard)
    const int Ncols = 32, rtpb = 8 / (Ncols >> 4);
    wmma_gemm_kernel<<<nblk(rowTiles, rtpb), 256, (size_t)FDIM * Ncols * sizeof(float), stream>>>(
        h, h, h, 1, Wm2, bm2, (float*)d_out, N, Ncols, 0);
  }
}